// PerVertQuaternion_12463995093943
// MI455X (gfx1250) — compile-verified
//
#include <hip/hip_runtime.h>
#include <math.h>

typedef __attribute__((ext_vector_type(2))) float v2f;
typedef __attribute__((ext_vector_type(8))) float v8f;

struct F3 { float x, y, z; };

__device__ __forceinline__ F3 f3(float x, float y, float z) { F3 r; r.x = x; r.y = y; r.z = z; return r; }
__device__ __forceinline__ F3 fsub(F3 a, F3 b) { return f3(a.x - b.x, a.y - b.y, a.z - b.z); }
__device__ __forceinline__ F3 fcross(F3 a, F3 b) {
    return f3(a.y * b.z - a.z * b.y,
              a.z * b.x - a.x * b.z,
              a.x * b.y - a.y * b.x);
}
__device__ __forceinline__ float fnorm(F3 a) { return sqrtf(a.x * a.x + a.y * a.y + a.z * a.z); }
__device__ __forceinline__ F3 fnormalize(F3 a) {
    float inv = 1.0f / fmaxf(fnorm(a), 1e-12f);
    return f3(a.x * inv, a.y * inv, a.z * inv);
}
__device__ __forceinline__ F3 loadv(const float* __restrict__ p, int idx) {
    return f3(p[3 * idx + 0], p[3 * idx + 1], p[3 * idx + 2]);
}
__device__ __forceinline__ float sqrt_pos(float x) { return x > 0.0f ? sqrtf(x) : 0.0f; }

// TBN frame of triangle (a,b,c): basis vectors X,Y,Z are the COLUMNS of R.
__device__ __forceinline__ void tbn(F3 a, F3 b, F3 c, F3& X, F3& Y, F3& Z) {
    F3 n = fnormalize(fcross(fsub(b, a), fsub(c, a)));
    F3 d = fsub(b, a);
    X = fnormalize(fcross(d, n));
    Y = fnormalize(fcross(d, X));
    Z = fnormalize(d);
}

// One thread per face. Per wave: 32 faces -> 8x V_WMMA_F32_16X16X4_F32
// (4 faces block-diagonally packed per 16x16 tile) computing P = R_d * R_c^T.
// Staged matrices are zero-padded to dense 4x4 tiles so every LDS fragment
// access is unconditional; fragment loads are software-pipelined one tile
// ahead so the WMMA waits need not drain the DS queue.
__global__ void __launch_bounds__(256)
pervert_quat_face_kernel(const float* __restrict__ mesh_verts,
                         const float* __restrict__ cano_verts,
                         const int*   __restrict__ cano_faces,
                         float*       __restrict__ vq,
                         int F) {
    __shared__ float4 sA4[8][32][4];   // per-wave padded A = R_d      (4x4, row i, col k)
    __shared__ float4 sB4[8][32][4];   // per-wave padded B = R_c^T    (4x4, row k, col j)
    __shared__ float4 sP4[8][32][3];   // per-wave P rows (3 rows, padded width 4)
    __shared__ float4 sDump[8][3];     // per-wave sink for non-owning-half stores

    const int tid   = blockIdx.x * 256 + threadIdx.x;
    const int wave  = threadIdx.x >> 5;
    const int lane  = threadIdx.x & 31;
    const bool valid = (tid < F);
    const int f = valid ? tid : (F - 1); // clamp: keeps EXEC all-ones for WMMA

    const int i0 = cano_faces[3 * f + 0];
    const int i1 = cano_faces[3 * f + 1];
    const int i2 = cano_faces[3 * f + 2];

    F3 ca = loadv(cano_verts, i0), cb = loadv(cano_verts, i1), cc = loadv(cano_verts, i2);
    F3 da = loadv(mesh_verts, i0), db = loadv(mesh_verts, i1), dc = loadv(mesh_verts, i2);

    // face area from canonical tri: |cross(c2-c1, c0-c1)| / 2
    const float area = 0.5f * fnorm(fcross(fsub(cc, cb), fsub(ca, cb)));

    F3 Xc, Yc, Zc, Xd, Yd, Zd;
    tbn(ca, cb, cc, Xc, Yc, Zc);
    tbn(da, db, dc, Xd, Yd, Zd);

    // A = R_d: A[i][k] = basis_d[k][i]; row/col 3 zero-padded.
    sA4[wave][lane][0] = make_float4(Xd.x, Yd.x, Zd.x, 0.0f);
    sA4[wave][lane][1] = make_float4(Xd.y, Yd.y, Zd.y, 0.0f);
    sA4[wave][lane][2] = make_float4(Xd.z, Yd.z, Zd.z, 0.0f);
    sA4[wave][lane][3] = make_float4(0.0f, 0.0f, 0.0f, 0.0f);
    // B = R_c^T: row k is the k-th canonical basis vector; row/col 3 zero-padded.
    sB4[wave][lane][0] = make_float4(Xc.x, Xc.y, Xc.z, 0.0f);
    sB4[wave][lane][1] = make_float4(Yc.x, Yc.y, Yc.z, 0.0f);
    sB4[wave][lane][2] = make_float4(Zc.x, Zc.y, Zc.z, 0.0f);
    sB4[wave][lane][3] = make_float4(0.0f, 0.0f, 0.0f, 0.0f);
    __syncthreads();

    // Fragment geometry for V_WMMA_F32_16X16X4_F32 (wave32):
    //  A 16x4: lanes 0-15 hold K={0,1}, lanes 16-31 hold K={2,3} of row (lane&15)
    //  B 4x16: lanes 0-15 hold K={0,1}, lanes 16-31 hold K={2,3} of col (lane&15)
    //  D 16x16: VGPR r = row r (lanes 0-15) / row r+8 (lanes 16-31), col = lane&15
    const int half = lane >> 4;           // which K-half / row-half
    const int rc   = lane & 15;           // A-row / B-col / D-col
    const int s    = rc >> 2;             // 4x4 diagonal slot 0..3
    const int ij   = rc & 3;              // index within slot
    const int k0   = half * 2;            // first K this lane holds
    const bool owns = (half == (s >> 1)); // this half holds rows 4s..4s+2

    auto loadA = [&](int t) -> v2f {
        const float* __restrict__ Af = (const float*)&sA4[wave][t * 4 + s][0];
        v2f a;
        a.x = Af[ij * 4 + k0];            // consecutive -> ds_load_b64
        a.y = Af[ij * 4 + k0 + 1];
        return a;
    };
    auto loadB = [&](int t) -> v2f {
        const float* __restrict__ Bf = (const float*)&sB4[wave][t * 4 + s][0];
        v2f b;
        b.x = Bf[k0 * 4 + ij];            // stride 16B -> ds_load_2addr_b32
        b.y = Bf[k0 * 4 + 4 + ij];
        return b;
    };

    v2f a = loadA(0);
    v2f b = loadB(0);
    #pragma unroll
    for (int t = 0; t < 8; ++t) {
        v2f an = a, bn = b;
        if (t < 7) {                      // prefetch next tile before this WMMA
            an = loadA(t + 1);
            bn = loadB(t + 1);
        }

        v8f cacc = {0.0f, 0.0f, 0.0f, 0.0f, 0.0f, 0.0f, 0.0f, 0.0f};
        v8f d = __builtin_amdgcn_wmma_f32_16x16x4_f32(
            /*neg_a=*/false, a, /*neg_b=*/false, b,
            /*c_mod=*/(short)0, cacc, /*reuse_a=*/false, /*reuse_b=*/false);

        // Diagonal block rows 4s..4s+2 of column rc belong to face g = t*4+s.
        // In-register row base = (s&1)*4. Non-owning half stores into the dump
        // row (address select, no exec divergence); ij==3 lands in padded .w.
        float e0, e1, e2;
        if (s & 1) { e0 = d[4]; e1 = d[5]; e2 = d[6]; }
        else       { e0 = d[0]; e1 = d[1]; e2 = d[2]; }
        float* pb = owns ? (float*)&sP4[wave][t * 4 + s][0] : (float*)&sDump[wave][0];
        pb[0 + ij] = e0;
        pb[4 + ij] = e1;
        pb[8 + ij] = e2;

        a = an;
        b = bn;
    }
    __syncthreads();

    const float4 r0 = sP4[wave][lane][0];
    const float4 r1 = sP4[wave][lane][1];
    const float4 r2 = sP4[wave][lane][2];
    const float m00 = r0.x, m01 = r0.y, m02 = r0.z;
    const float m10 = r1.x, m11 = r1.y, m12 = r1.z;
    const float m20 = r2.x, m21 = r2.y, m22 = r2.z;

    const float q0 = sqrt_pos(1.0f + m00 + m11 + m22);
    const float q1 = sqrt_pos(1.0f + m00 - m11 - m22);
    const float q2 = sqrt_pos(1.0f - m00 + m11 - m22);
    const float q3 = sqrt_pos(1.0f - m00 - m11 + m22);

    int   idx  = 0;
    float best = q0;
    if (q1 > best) { best = q1; idx = 1; }
    if (q2 > best) { best = q2; idx = 2; }
    if (q3 > best) { best = q3; idx = 3; }

    const float scale = area * (0.5f / fmaxf(best, 0.1f));
    float w0, w1, w2, w3;
    if (idx == 0)      { w0 = q0 * q0;   w1 = m21 - m12; w2 = m02 - m20; w3 = m10 - m01; }
    else if (idx == 1) { w0 = m21 - m12; w1 = q1 * q1;   w2 = m10 + m01; w3 = m02 + m20; }
    else if (idx == 2) { w0 = m02 - m20; w1 = m10 + m01; w2 = q2 * q2;   w3 = m12 + m21; }
    else               { w0 = m10 - m01; w1 = m20 + m02; w2 = m21 + m12; w3 = q3 * q3;   }
    w0 *= scale; w1 *= scale; w2 *= scale; w3 *= scale;

    if (valid) {
        atomicAdd(&vq[4 * i0 + 0], w0); atomicAdd(&vq[4 * i0 + 1], w1);
        atomicAdd(&vq[4 * i0 + 2], w2); atomicAdd(&vq[4 * i0 + 3], w3);
        atomicAdd(&vq[4 * i1 + 0], w0); atomicAdd(&vq[4 * i1 + 1], w1);
        atomicAdd(&vq[4 * i1 + 2], w2); atomicAdd(&vq[4 * i1 + 3], w3);
        atomicAdd(&vq[4 * i2 + 0], w0); atomicAdd(&vq[4 * i2 + 1], w1);
        atomicAdd(&vq[4 * i2 + 2], w2); atomicAdd(&vq[4 * i2 + 3], w3);
    }
}

__global__ void __launch_bounds__(256)
pervert_quat_normalize_kernel(const float4* __restrict__ vq,
                              float4* __restrict__ out, int V) {
    const int v = blockIdx.x * 256 + threadIdx.x;
    if (v >= V) return;
    const float4 q = vq[v];
    const float inv = 1.0f / fmaxf(sqrtf(q.x * q.x + q.y * q.y + q.z * q.z + q.w * q.w), 1e-6f);
    out[v] = make_float4(q.x * inv, q.y * inv, q.z * inv, q.w * inv);
}

extern "C" void kernel_launch(void* const* d_in, const int* in_sizes, int n_in,
                              void* d_out, int out_size, void* d_ws, size_t ws_size,
                              hipStream_t stream) {
    const float* mesh_verts = (const float*)d_in[0];
    const float* cano_verts = (const float*)d_in[1];
    const int*   cano_faces = (const int*)d_in[2];

    const int V = in_sizes[0] / 3;
    const int F = in_sizes[2] / 3;

    float* vq = (float*)d_ws; // V*4 f32 accumulator (4 MB) — L2-resident
    hipMemsetAsync(vq, 0, (size_t)V * 4 * sizeof(float), stream);

    const int fblocks = (F + 255) / 256;
    pervert_quat_face_kernel<<<fblocks, 256, 0, stream>>>(
        mesh_verts, cano_verts, cano_faces, vq, F);

    const int vblocks = (V + 255) / 256;
    pervert_quat_normalize_kernel<<<vblocks, 256, 0, stream>>>(
        (const float4*)vq, (float4*)d_out, V);
}